// AudioGSModel_24017457119798
// MI455X (gfx1250) — compile-verified
//
#include <hip/hip_runtime.h>
#include <math.h>

typedef __attribute__((ext_vector_type(2))) float v2f;
typedef __attribute__((ext_vector_type(8))) float v8f;

#define MAX_ATOMS 2048
#define SUBGRP    (MAX_ATOMS / 4)   // 512 atoms per K-subgroup
#define NYQ       12000.0f
#define TWO_PI    6.2831853071795864769f

__device__ __forceinline__ float softplus_f(float x) {
    return (x > 20.0f) ? x : log1pf(__expf(x));
}

// ---------------------------------------------------------------------------
// Kernel 1: per-atom parameter transforms -> packed AoS in workspace.
// Layout per atom (8 floats, 32B):
//   amp, w0 = 2*pi*omega, c2 = -0.5/sigma^2, phi, tau, gamma, 0, 0
// c2 pre-folds the Gaussian so the hot loop needs q = dt^2 only once for
// both the envelope argument (c2*q) and the chirp phase (gamma*q).
// ---------------------------------------------------------------------------
__global__ void gabor_prep_kernel(const float* __restrict__ amplitude_logit,
                                  const float* __restrict__ tau,
                                  const float* __restrict__ omega_logit,
                                  const float* __restrict__ sigma_logit,
                                  const float* __restrict__ phi_vector,
                                  const float* __restrict__ gamma,
                                  float* __restrict__ atoms, int N) {
    int i = blockIdx.x * blockDim.x + threadIdx.x;
    if (i >= N) return;
    float amp   = softplus_f(amplitude_logit[i]);
    float omega = (0.99f * NYQ) / (1.0f + __expf(-omega_logit[i]));   // sigmoid * 0.99 * nyquist
    float sigma = softplus_f(sigma_logit[i]) + 1.0e-3f;
    float inv_s = 1.0f / sigma;
    float phi   = atan2f(phi_vector[2 * i + 1], phi_vector[2 * i]);
    float* p = atoms + 8 * i;
    p[0] = amp;
    p[1] = TWO_PI * omega;
    p[2] = -0.5f * inv_s * inv_s;
    p[3] = phi;
    p[4] = tau[i];
    p[5] = gamma[i];
    p[6] = 0.0f;
    p[7] = 0.0f;
}

// ---------------------------------------------------------------------------
// Kernel 2: additive synthesis.
// Block = 256 threads = 8 waves. Each wave owns 16 consecutive time samples.
// A-matrix (16x4 f32) per ISA layout:
//   lanes 0-15 : M = lane,     K=0 (acc0 -> A vgpr0), K=1 (acc1 -> A vgpr1)
//   lanes 16-31: M = lane-16,  K=2 (acc0),            K=3 (acc1)
// Atoms partitioned into 4 contiguous subgroups of 512 along K.
// One v_wmma_f32_16x16x4_f32 with B = ones reduces the 4 partials per sample.
// ---------------------------------------------------------------------------
__global__ void __launch_bounds__(256)
gabor_synth_kernel(const float* __restrict__ atoms,
                   const float* __restrict__ tvec,
                   float* __restrict__ out, int N, int T) {
    __shared__ float4 lds[MAX_ATOMS * 2];   // 64 KB: two float4 per atom

    const int tid = threadIdx.x;

    // Stage all atom params into LDS (coalesced b128 loads/stores).
    for (int a = tid; a < MAX_ATOMS; a += 256) {
        if (a < N) {
            const float4* src = (const float4*)(atoms + 8 * a);
            lds[2 * a]     = src[0];
            lds[2 * a + 1] = src[1];
        } else {
            lds[2 * a]     = make_float4(0.0f, 0.0f, 0.0f, 0.0f); // amp=0 -> contributes 0
            lds[2 * a + 1] = make_float4(0.0f, 0.0f, 0.0f, 0.0f);
        }
    }
    __syncthreads();

    const int wave  = tid >> 5;        // 0..7
    const int lane  = tid & 31;
    const int m     = lane & 15;       // M index (time row within the wave tile)
    const int half  = lane >> 4;       // selects K-pair {0,1} or {2,3}
    const int tbase = blockIdx.x * 128 + wave * 16;

    int tidx = tbase + m;
    if (tidx >= T) tidx = T - 1;       // tail clamp (no-op for T=48000); keeps EXEC full
    const float t = tvec[tidx];

    const int base = (2 * half) * SUBGRP;   // start of this lane's first K-subgroup

    float acc0 = 0.0f;   // K = 2*half
    float acc1 = 0.0f;   // K = 2*half + 1

    #pragma unroll 4
    for (int i = 0; i < SUBGRP; ++i) {
        {
            const int a = base + i;
            float4 p0 = lds[2 * a];
            float4 p1 = lds[2 * a + 1];
            float dt  = t - p1.x;                           // t - tau
            float q   = dt * dt;
            float env = __expf(p0.z * q);                   // exp(-0.5*(dt/sigma)^2)
            float ph  = fmaf(p1.y, q, fmaf(p0.y, dt, p0.w));// gamma*dt^2 + w0*dt + phi
            acc0 = fmaf(p0.x * env, __cosf(ph), acc0);
        }
        {
            const int a = base + SUBGRP + i;
            float4 p0 = lds[2 * a];
            float4 p1 = lds[2 * a + 1];
            float dt  = t - p1.x;
            float q   = dt * dt;
            float env = __expf(p0.z * q);
            float ph  = fmaf(p1.y, q, fmaf(p0.y, dt, p0.w));
            acc1 = fmaf(p0.x * env, __cosf(ph), acc1);
        }
    }

    // Cross-subgroup reduction: D[m,n] = sum_k A[m,k] * 1 + 0
    v2f A; A.x = acc0; A.y = acc1;
    v2f B; B.x = 1.0f; B.y = 1.0f;
    v8f C = {};
    v8f D = __builtin_amdgcn_wmma_f32_16x16x4_f32(
        /*neg_a=*/false, A, /*neg_b=*/false, B,
        /*c_mod=*/(short)0, C, /*reuse_a=*/false, /*reuse_b=*/false);

    // C/D layout: lanes 0-15 hold M=0..7 (N=lane), lanes 16-31 hold M=8..15.
    // Column N=0 lanes (lane 0 and lane 16) store 8 contiguous results each.
    if (m == 0) {
        const int ob = tbase + 8 * half;
        if (ob + 8 <= T) {
            float4 lo = make_float4(D[0], D[1], D[2], D[3]);
            float4 hi = make_float4(D[4], D[5], D[6], D[7]);
            *(float4*)(out + ob)     = lo;
            *(float4*)(out + ob + 4) = hi;
        } else if (ob < T) {
            // cold generic tail (never taken for T % 128 == 0)
            float tmp[8] = {D[0], D[1], D[2], D[3], D[4], D[5], D[6], D[7]};
            int nrem = T - ob;
            for (int j = 0; j < nrem; ++j) out[ob + j] = tmp[j];
        }
    }
}

// ---------------------------------------------------------------------------
extern "C" void kernel_launch(void* const* d_in, const int* in_sizes, int n_in,
                              void* d_out, int out_size, void* d_ws, size_t ws_size,
                              hipStream_t stream) {
    const float* amplitude_logit = (const float*)d_in[0];
    const float* tau             = (const float*)d_in[1];
    const float* omega_logit     = (const float*)d_in[2];
    const float* sigma_logit     = (const float*)d_in[3];
    const float* phi_vector      = (const float*)d_in[4];
    const float* gamma           = (const float*)d_in[5];
    const float* tvec            = (const float*)d_in[6];

    const int N = in_sizes[0];     // 2048 atoms
    const int T = in_sizes[6];     // 48000 samples
    float* out   = (float*)d_out;
    float* atoms = (float*)d_ws;   // 8 floats per atom (needs 8*N*4 = 64 KB)

    int prep_blocks = (N + 255) / 256;
    gabor_prep_kernel<<<prep_blocks, 256, 0, stream>>>(
        amplitude_logit, tau, omega_logit, sigma_logit, phi_vector, gamma, atoms, N);

    int synth_blocks = (T + 127) / 128;   // 128 samples per block (8 waves x 16)
    gabor_synth_kernel<<<synth_blocks, 256, 0, stream>>>(atoms, tvec, out, N, T);
}